// GSC_YOLO_34986803593858
// MI455X (gfx1250) — compile-verified
//
#include <hip/hip_runtime.h>
#include <hip/hip_bf16.h>
#include <math.h>

#define IOU_THR     0.5f
#define SIGMA_INV   5.0f          /* 1/0.2 */
#define BETA_INV    (1.0f/0.6f)
#define EPS_F       1e-8f
#define SCORE_FLOOR 0.01f

typedef __attribute__((ext_vector_type(2))) float v2f;
typedef __attribute__((ext_vector_type(8))) float v8f;

__device__ __forceinline__ v8f wmma4(v2f a, v2f b, v8f c) {
  // D = A(16x4) * B(4x16) + C, f32, wave32
  return __builtin_amdgcn_wmma_f32_16x16x4_f32(false, a, false, b, (short)0, c,
                                               false, false);
}

// ---------------------------------------------------------------------------
// Kernel 1: stable descending argsort by counting; gather sorted arrays.
// rank[i] = #{j: s[j] > s[i]} + #{j < i: s[j] == s[i]}  (matches argsort(-s))
// ---------------------------------------------------------------------------
__global__ void rank_sort_kernel(const float* __restrict__ boxes,
                                 const float* __restrict__ scores,
                                 float* __restrict__ boxes_s,
                                 float* __restrict__ scores_s,
                                 int* __restrict__ order, int n) {
  int i = blockIdx.x * blockDim.x + threadIdx.x;
  if (i >= n) return;
  float si = scores[i];
  int rank = 0;
  for (int j = 0; j < n; ++j) {
    float sj = scores[j];
    rank += (sj > si) || ((sj == si) && (j < i));
  }
  boxes_s[4 * rank + 0] = boxes[4 * i + 0];
  boxes_s[4 * rank + 1] = boxes[4 * i + 1];
  boxes_s[4 * rank + 2] = boxes[4 * i + 2];
  boxes_s[4 * rank + 3] = boxes[4 * i + 3];
  scores_s[rank] = si;
  order[rank] = i;
}

// ---------------------------------------------------------------------------
// Kernel 2: pseudo[i][k] = IoU(sorted_box i, sorted_box k), k = 0..3
// ---------------------------------------------------------------------------
__global__ void pseudo_kernel(const float* __restrict__ bs,
                              float* __restrict__ pseudo, int n) {
  int i = blockIdx.x * blockDim.x + threadIdx.x;
  if (i >= n) return;
  float b0 = bs[4 * i + 0], b1 = bs[4 * i + 1];
  float b2 = bs[4 * i + 2], b3 = bs[4 * i + 3];
  float ai = (b2 - b0) * (b3 - b1);
#pragma unroll
  for (int k = 0; k < 4; ++k) {
    float c0 = bs[4 * k + 0], c1 = bs[4 * k + 1];
    float c2 = bs[4 * k + 2], c3 = bs[4 * k + 3];
    float iw = fminf(b2, c2) - fmaxf(b0, c0);
    float ih = fminf(b3, c3) - fmaxf(b1, c1);
    float inter = fmaxf(iw, 0.f) * fmaxf(ih, 0.f);
    float ak = (c2 - c0) * (c3 - c1);
    pseudo[4 * i + k] = inter / (ai + ak - inter + EPS_F);
  }
}

// ---------------------------------------------------------------------------
// Kernel 3: blocked sequential scan (single workgroup, state in LDS).
// Dynamic LDS layout: sc[n] f32 | keep[n] u32 | rbox[64] | rps[64] | rkf[16]
// ---------------------------------------------------------------------------
__global__ __launch_bounds__(1024) void nms_scan_kernel(
    const float* __restrict__ bs, const float* __restrict__ ss,
    const float* __restrict__ ps, const int* __restrict__ order,
    float* __restrict__ out, int n) {
  extern __shared__ unsigned char smem[];
  float* sc_s = (float*)smem;
  unsigned* kp_s = (unsigned*)(sc_s + n);
  float* rbox = (float*)(kp_s + n);
  float* rps = rbox + 64;
  float* rkf = rps + 64;

  const int tid = threadIdx.x;
  const int lane = tid & 31;
  const bool hi = (lane >= 16);
  const int mr = lane & 15;
  const int wave = tid >> 5;

  for (int r = tid; r < n; r += 1024) { sc_s[r] = ss[r]; kp_s[r] = 1u; }
  __syncthreads();

  const int nb = n >> 4;
  for (int blk = 0; blk < nb; ++blk) {
    const int i0 = blk << 4;
    // ---- stage the 16 row boxes / pseudo rows ----
    if (tid < 64) {
      rbox[tid] = bs[(size_t)i0 * 4 + tid];
      rps[tid] = ps[(size_t)i0 * 4 + tid];
    }
    __syncthreads();

    // ---- Phase A: sequential resolution of the 16x16 diagonal tile ----
    if (tid < 16) {
      int j = i0 + tid;
      float scj = sc_s[j];
      float kj = kp_s[j] ? 1.0f : 0.0f;
      float b0 = rbox[4 * tid + 0], b1 = rbox[4 * tid + 1];
      float b2 = rbox[4 * tid + 2], b3 = rbox[4 * tid + 3];
      float q0 = rps[4 * tid + 0], q1 = rps[4 * tid + 1];
      float q2 = rps[4 * tid + 2], q3 = rps[4 * tid + 3];
      float aj = (b2 - b0) * (b3 - b1);
      float apj = (q2 - q0) * (q3 - q1);
      float cxj = 0.5f * (q0 + q2), cyj = 0.5f * (q1 + q3);
      for (int i = 0; i < 15; ++i) {
        float ki = __shfl(kj, i);  // row i's live keep (lockstep in wave)
        float r0 = rbox[4 * i + 0], r1 = rbox[4 * i + 1];
        float r2 = rbox[4 * i + 2], r3 = rbox[4 * i + 3];
        float iw = fminf(r2, b2) - fmaxf(r0, b0);
        float ih = fminf(r3, b3) - fmaxf(r1, b1);
        float inter = fmaxf(iw, 0.f) * fmaxf(ih, 0.f);
        float ar = (r2 - r0) * (r3 - r1);
        float iou = inter / (ar + aj - inter + EPS_F);
        float p0 = rps[4 * i + 0], p1 = rps[4 * i + 1];
        float p2 = rps[4 * i + 2], p3 = rps[4 * i + 3];
        float pw = fminf(p2, q2) - fmaxf(p0, q0);
        float ph = fminf(p3, q3) - fmaxf(p1, q1);
        float interp = fmaxf(pw, 0.f) * fmaxf(ph, 0.f);
        float apr = (p2 - p0) * (p3 - p1);
        float ioup = interp / (apr + apj - interp + EPS_F);
        float cxr = 0.5f * (p0 + p2), cyr = 0.5f * (p1 + p3);
        float dx = cxr - cxj, dy = cyr - cyj;
        float cdist = dx * dx + dy * dy;
        float ew = fmaxf(p2, q2) - fminf(p0, q0);
        float eh = fmaxf(p3, q3) - fminf(p1, q1);
        float edist = ew * ew + eh * eh;
        float diou = ioup - cdist / (edist + EPS_F);
        float pen = expf(-iou * iou * SIGMA_INV);
        float tot = fminf(pen + diou * diou * BETA_INV, 1.0f);
        bool act = (tid > i) && (ki != 0.f) && (iou > IOU_THR) && (kj != 0.f);
        float ns = act ? scj * tot : scj;
        kj = (act && (ns < SCORE_FLOOR)) ? 0.f : kj;
        scj = ns;
      }
      sc_s[j] = scj;
      kp_s[j] = (kj != 0.f) ? 1u : 0u;
      rkf[tid] = kj;
    }
    __syncthreads();

    // ---- row-keep mask for this block (wave-uniform) ----
    unsigned rmask = 0;
#pragma unroll
    for (int i = 0; i < 16; ++i) rmask |= (rkf[i] != 0.f ? 1u : 0u) << i;

    const int c0base = i0 + 16;
    const int ntiles = (n - c0base) >> 4;
    if (rmask != 0u) {
      // ---- per-lane row features for WMMA A operands (block-invariant) ----
      float r0 = rbox[4 * mr + 0], r1 = rbox[4 * mr + 1];
      float r2 = rbox[4 * mr + 2], r3 = rbox[4 * mr + 3];
      float ar = (r2 - r0) * (r3 - r1);
      float p0r = rps[4 * mr + 0], p1r = rps[4 * mr + 1];
      float p2r = rps[4 * mr + 2], p3r = rps[4 * mr + 3];
      float apr = (p2r - p0r) * (p3r - p1r);
      float cxr = 0.5f * (p0r + p2r), cyr = 0.5f * (p1r + p3r);
      float s2r = cxr * cxr + cyr * cyr;
      v2f Acd, Asb, Asp;
      Acd.x = hi ? (-2.f * cyr) : s2r;  Acd.y = hi ? 1.f : (-2.f * cxr);
      Asb.x = hi ? 0.f : ar;            Asb.y = hi ? 0.f : 1.f;
      Asp.x = hi ? 0.f : apr;           Asp.y = hi ? 0.f : 1.f;

      // ---- Phase B: apply 16 finalized rows to all later columns ----
      for (int t = wave; t < ntiles; t += 32) {
        int j = c0base + (t << 4) + mr;
        float b0 = bs[4 * j + 0], b1 = bs[4 * j + 1];
        float b2 = bs[4 * j + 2], b3 = bs[4 * j + 3];
        float q0 = ps[4 * j + 0], q1 = ps[4 * j + 1];
        float q2 = ps[4 * j + 2], q3 = ps[4 * j + 3];
        float aj = (b2 - b0) * (b3 - b1);
        float apj = (q2 - q0) * (q3 - q1);
        float cxj = 0.5f * (q0 + q2), cyj = 0.5f * (q1 + q3);
        float s2c = cxj * cxj + cyj * cyj;
        v2f Bcd, Bsb, Bsp;
        Bcd.x = hi ? cyj : 1.f;  Bcd.y = hi ? s2c : cxj;
        Bsb.x = hi ? 0.f : 1.f;  Bsb.y = hi ? 0.f : aj;
        Bsp.x = hi ? 0.f : 1.f;  Bsp.y = hi ? 0.f : apj;

        v8f cd = {}; cd = wmma4(Acd, Bcd, cd);  // center-dist matrix
        v8f sb = {}; sb = wmma4(Asb, Bsb, sb);  // area_i + area_j (boxes)
        v8f sp = {}; sp = wmma4(Asp, Bsp, sp);  // area_i + area_j (pseudo)

        // reunify each column's 16 rows per lane (halves live in lane^16)
        float cdv[16], sbv[16], spv[16];
#pragma unroll
        for (int k = 0; k < 8; ++k) {
          float o1 = __shfl_xor(cd[k], 16);
          float o2 = __shfl_xor(sb[k], 16);
          float o3 = __shfl_xor(sp[k], 16);
          cdv[k] = hi ? o1 : cd[k];  cdv[k + 8] = hi ? cd[k] : o1;
          sbv[k] = hi ? o2 : sb[k];  sbv[k + 8] = hi ? sb[k] : o2;
          spv[k] = hi ? o3 : sp[k];  spv[k + 8] = hi ? sp[k] : o3;
        }

        float scj = sc_s[j];
        float kj = kp_s[j] ? 1.0f : 0.0f;
#pragma unroll
        for (int i = 0; i < 16; ++i) {
          float x0 = rbox[4 * i + 0], x1 = rbox[4 * i + 1];
          float x2 = rbox[4 * i + 2], x3 = rbox[4 * i + 3];
          float iw = fminf(x2, b2) - fmaxf(x0, b0);
          float ih = fminf(x3, b3) - fmaxf(x1, b1);
          float inter = fmaxf(iw, 0.f) * fmaxf(ih, 0.f);
          float iou = inter / (sbv[i] - inter + EPS_F);
          float p0 = rps[4 * i + 0], p1 = rps[4 * i + 1];
          float p2 = rps[4 * i + 2], p3 = rps[4 * i + 3];
          float pw = fminf(p2, q2) - fmaxf(p0, q0);
          float ph = fminf(p3, q3) - fmaxf(p1, q1);
          float interp = fmaxf(pw, 0.f) * fmaxf(ph, 0.f);
          float ioup = interp / (spv[i] - interp + EPS_F);
          float ew = fmaxf(p2, q2) - fminf(p0, q0);
          float eh = fmaxf(p3, q3) - fminf(p1, q1);
          float edist = ew * ew + eh * eh;
          float diou = ioup - cdv[i] / (edist + EPS_F);
          float pen = expf(-iou * iou * SIGMA_INV);
          float tot = fminf(pen + diou * diou * BETA_INV, 1.0f);
          bool act = ((rmask >> i) & 1u) && (iou > IOU_THR) && (kj != 0.f);
          float ns = act ? scj * tot : scj;
          kj = (act && (ns < SCORE_FLOOR)) ? 0.f : kj;
          scj = ns;
        }
        if (!hi) {
          sc_s[j] = scj;
          kp_s[j] = (kj != 0.f) ? 1u : 0u;
        }
      }
    }
    __syncthreads();
  }

  // ---- scatter back to original order: [scores | keep] ----
  for (int r = tid; r < n; r += 1024) {
    int oi = order[r];
    out[oi] = sc_s[r];
    out[n + oi] = kp_s[r] ? 1.0f : 0.0f;
  }
}

// ---------------------------------------------------------------------------
extern "C" void kernel_launch(void* const* d_in, const int* in_sizes, int n_in,
                              void* d_out, int out_size, void* d_ws,
                              size_t ws_size, hipStream_t stream) {
  const float* boxes = (const float*)d_in[0];
  const float* scores = (const float*)d_in[1];
  const int n = in_sizes[1];

  float* ws = (float*)d_ws;
  float* boxes_s = ws;                       // n*4
  float* scores_s = ws + (size_t)4 * n;      // n
  float* pseudo = ws + (size_t)5 * n;        // n*4
  int* order = (int*)(ws + (size_t)9 * n);   // n

  const int threads = 256;
  const int blocks = (n + threads - 1) / threads;
  rank_sort_kernel<<<blocks, threads, 0, stream>>>(boxes, scores, boxes_s,
                                                   scores_s, order, n);
  pseudo_kernel<<<blocks, threads, 0, stream>>>(boxes_s, pseudo, n);

  size_t shmem = (size_t)n * 8 + (64 + 64 + 16) * sizeof(float);
  nms_scan_kernel<<<1, 1024, shmem, stream>>>(boxes_s, scores_s, pseudo, order,
                                              (float*)d_out, n);
}